// TensorProcessor_8435315770049
// MI455X (gfx1250) — compile-verified
//
#include <hip/hip_runtime.h>
#include <hip/hip_bf16.h>

typedef __attribute__((ext_vector_type(2))) float v2f;
typedef __attribute__((ext_vector_type(8))) float v8f;

#define Y_ELEMS   (32*100*128*64)   // 26,214,400
#define BUF_ELEMS (8*100*128*64)    // 6,553,600 (largest intermediate: [8,100,128,64])
#define GRAM_WAVES 8                // 256 threads / block

// ---------------- Conv3d(1x3x1) + bias + ReLU ----------------
// y[c,d,h,w] = relu(b[c] + sum_{i<4,kh<3} x[i,d,h+kh,w] * w[c,i,0,kh,0])
__global__ void conv_relu_kernel(const float* __restrict__ x,
                                 const float* __restrict__ w,
                                 const float* __restrict__ b,
                                 float* __restrict__ y) {
  int tid = blockIdx.x * blockDim.x + threadIdx.x;
  if (tid >= Y_ELEMS) return;
  int wi = tid & 63;
  int h  = (tid >> 6) & 127;
  int dd = (tid >> 13) % 100;
  int c  = tid / (100 * 128 * 64);
  float acc = b[c];
  const float* wc = w + c * 12;
#pragma unroll
  for (int i = 0; i < 4; ++i) {
    const float* xi = x + (((long)i * 100 + dd) * 130 + h) * 64 + wi;
#pragma unroll
    for (int kh = 0; kh < 3; ++kh)
      acc += xi[kh * 64] * wc[i * 3 + kh];
  }
  y[tid] = acc > 0.f ? acc : 0.f;
}

__global__ void zero_kernel(float* __restrict__ p, int n) {
  int tid = blockIdx.x * blockDim.x + threadIdx.x;
  if (tid < n) p[tid] = 0.f;
}

// ---------------- Gram matrix G += X_(m) X_(m)^T via fp32 WMMA ----------------
// Element X(i,k): k -> (o = k/Dinner, in = k%Dinner); addr = (o*d + i)*Dinner + in.
// Column order is irrelevant for a Gram matrix, so no unfolding materialization.
// Grid: (tiles_i, tiles_j, k_chunks). Block = 8 waves; each wave owns interleaved
// 4-wide K slices of this block's chunk; LDS reduce; partial tile atomically
// accumulated into pre-zeroed G (K-split gives enough blocks to stream at HBM/L2 rate).
__global__ void gram_wmma_kernel(const float* __restrict__ T, float* __restrict__ G,
                                 int d, int Dinner, int N, int chunk) {
  int lane = threadIdx.x & 31;
  int wave = threadIdx.x >> 5;
  int ti = blockIdx.x, tj = blockIdx.y;
  int kbeg = blockIdx.z * chunk;
  int kend = kbeg + chunk; if (kend > N) kend = N;

  int mrow = ti * 16 + (lane & 15);          // A row
  int ncol = tj * 16 + (lane & 15);          // B col (= A row on the transpose side)
  int koff = (lane < 16) ? 0 : 2;            // f32 A/B K-pair base per half-wave

  v8f c = {0.f, 0.f, 0.f, 0.f, 0.f, 0.f, 0.f, 0.f};

  for (int k0 = kbeg + wave * 4; k0 < kend; k0 += GRAM_WAVES * 4) {
    // prefetch a future A element (emits global_prefetch_b8)
    int kp = k0 + 128;
    if (kp < N && mrow < d) {
      long pa;
      if (Dinner == 1) pa = (long)kp * d + mrow;
      else {
        int op = kp / Dinner, inp = kp - op * Dinner;
        pa = ((long)op * d + mrow) * Dinner + inp;
      }
      __builtin_prefetch(&T[pa], 0, 1);
    }
    v2f a, bv;
#pragma unroll
    for (int e = 0; e < 2; ++e) {
      int k = k0 + koff + e;
      float av = 0.f, bb = 0.f;
      if (k < kend) {
        long base;
        if (Dinner == 1) {
          base = (long)k * d;
          if (mrow < d) av = T[base + mrow];
          if (ncol < d) bb = T[base + ncol];
        } else {
          int o = k / Dinner, in = k - o * Dinner;
          base = (long)o * d * Dinner + in;
          if (mrow < d) av = T[base + (long)mrow * Dinner];
          if (ncol < d) bb = T[base + (long)ncol * Dinner];
        }
      }
      a[e] = av;
      bv[e] = bb;
    }
    c = __builtin_amdgcn_wmma_f32_16x16x4_f32(false, a, false, bv,
                                              (short)0, c, false, false);
  }

  __shared__ float red[GRAM_WAVES][32][8];
#pragma unroll
  for (int j = 0; j < 8; ++j) red[wave][lane][j] = c[j];
  __syncthreads();

  if (wave == 0) {
#pragma unroll
    for (int j = 0; j < 8; ++j) {
      float v = 0.f;
#pragma unroll
      for (int ww = 0; ww < GRAM_WAVES; ++ww) v += red[ww][lane][j];
      int M  = ti * 16 + j + ((lane < 16) ? 0 : 8);
      int Nc = tj * 16 + (lane & 15);
      if (M < d && Nc < d) atomicAdd(&G[M * d + Nc], v);
    }
  }
}

// ---------------- top-r invariant subspace of symmetric PSD G (d<=128, r<=8) ----
// Orthogonal iteration: Q <- MGS(G*(G*Q)), repeated. Converges to the top-r
// eigenvector basis (descending eigenvalues) = top-r left singular vectors.
__global__ void subspace_eigen_kernel(const float* __restrict__ G,
                                      float* __restrict__ U, int d, int r) {
  __shared__ float Q[128 * 8];
  __shared__ float Z[128 * 8];
  __shared__ float red[128];
  int tid = threadIdx.x;

  if (tid < d) {
    for (int j = 0; j < 8; ++j) {
      unsigned h = (unsigned)(tid * 8 + j + 1) * 2654435761u;
      h ^= h >> 16; h *= 2246822519u; h ^= h >> 13;
      Q[tid * 8 + j] = ((float)(h & 0xFFFF) / 65536.0f) - 0.5f;
    }
  }
  __syncthreads();

  for (int it = 0; it < 16; ++it) {
    // Z = G * Q
    if (tid < d) {
      float acc[8] = {0.f, 0.f, 0.f, 0.f, 0.f, 0.f, 0.f, 0.f};
      for (int k = 0; k < d; ++k) {
        float g = G[tid * d + k];
        for (int j = 0; j < r; ++j) acc[j] += g * Q[k * 8 + j];
      }
      for (int j = 0; j < r; ++j) Z[tid * 8 + j] = acc[j];
    }
    __syncthreads();
    // Q = G * Z   (second power step)
    if (tid < d) {
      float acc[8] = {0.f, 0.f, 0.f, 0.f, 0.f, 0.f, 0.f, 0.f};
      for (int k = 0; k < d; ++k) {
        float g = G[tid * d + k];
        for (int j = 0; j < r; ++j) acc[j] += g * Z[k * 8 + j];
      }
      for (int j = 0; j < r; ++j) Q[tid * 8 + j] = acc[j];
    }
    __syncthreads();
    // Modified Gram-Schmidt on columns 0..r-1 of Q
    for (int j = 0; j < r; ++j) {
      for (int p = 0; p < j; ++p) {
        red[tid] = (tid < d) ? Q[tid * 8 + p] * Q[tid * 8 + j] : 0.f;
        __syncthreads();
        for (int s = 64; s > 0; s >>= 1) {
          if (tid < s) red[tid] += red[tid + s];
          __syncthreads();
        }
        float dot = red[0];
        __syncthreads();
        if (tid < d) Q[tid * 8 + j] -= dot * Q[tid * 8 + p];
        __syncthreads();
      }
      red[tid] = (tid < d) ? Q[tid * 8 + j] * Q[tid * 8 + j] : 0.f;
      __syncthreads();
      for (int s = 64; s > 0; s >>= 1) {
        if (tid < s) red[tid] += red[tid + s];
        __syncthreads();
      }
      float inv = rsqrtf(red[0] + 1e-30f);
      __syncthreads();
      if (tid < d) Q[tid * 8 + j] *= inv;
      __syncthreads();
    }
  }

  if (tid < d)
    for (int j = 0; j < 8; ++j)
      U[tid * 8 + j] = (j < r) ? Q[tid * 8 + j] : 0.f;
}

// ---------------- mode-k product: out[o,r,in] = sum_i U[i][r] * t[o,i,in] -------
__global__ void mode_dot_kernel(const float* __restrict__ T, const float* __restrict__ U,
                                float* __restrict__ Out, int Dk, int Dinner, int R,
                                int total) {
  int tid = blockIdx.x * blockDim.x + threadIdx.x;
  if (tid >= total) return;
  int in = tid % Dinner;
  int r  = (tid / Dinner) % R;
  int o  = tid / (Dinner * R);
  const float* tp = T + (long)o * Dk * Dinner + in;
  float acc = 0.f;
  for (int i = 0; i < Dk; ++i)
    acc += U[i * 8 + r] * tp[(long)i * Dinner];
  Out[tid] = acc;
}

extern "C" void kernel_launch(void* const* d_in, const int* in_sizes, int n_in,
                              void* d_out, int out_size, void* d_ws, size_t ws_size,
                              hipStream_t stream) {
  const float* x  = (const float*)d_in[0];
  const float* cw = (const float*)d_in[1];
  const float* cb = (const float*)d_in[2];
  float* out = (float*)d_out;
  float* ws  = (float*)d_ws;

  float* Y  = ws;
  float* BA = Y + Y_ELEMS;
  float* BB = BA + BUF_ELEMS;
  float* G  = BB + BUF_ELEMS;
  float* Ufac[4];
  Ufac[0] = G + 128 * 128;
  Ufac[1] = Ufac[0] + 32 * 8;
  Ufac[2] = Ufac[1] + 100 * 8;
  Ufac[3] = Ufac[2] + 128 * 8;

  const int ranks[4] = {8, 6, 6, 6};

  conv_relu_kernel<<<(Y_ELEMS + 255) / 256, 256, 0, stream>>>(x, cw, cb, Y);

  auto gram = [&](const float* T, const int* dims, int m) {
    int d = dims[m];
    int Dinner = 1; for (int i = m + 1; i < 4; ++i) Dinner *= dims[i];
    int Douter = 1; for (int i = 0; i < m; ++i) Douter *= dims[i];
    int N = Douter * Dinner;
    // split K across gridDim.z so big Grams have enough blocks to stream L2/HBM
    int kz = N / 8192; if (kz < 1) kz = 1; if (kz > 64) kz = 64;
    int chunk = ((N + kz * 16 - 1) / (kz * 16)) * 16;   // multiple of 16
    kz = (N + chunk - 1) / chunk;
    zero_kernel<<<(d * d + 255) / 256, 256, 0, stream>>>(G, d * d);
    dim3 grid((d + 15) / 16, (d + 15) / 16, kz);
    gram_wmma_kernel<<<grid, GRAM_WAVES * 32, 0, stream>>>(T, G, d, Dinner, N, chunk);
  };

  auto eig = [&](int m, int d) {
    subspace_eigen_kernel<<<1, 128, 0, stream>>>(G, Ufac[m], d, ranks[m]);
  };

  auto modedot = [&](const float* T, int* dims, int k, float* Out) {
    int Dk = dims[k];
    int Dinner = 1; for (int i = k + 1; i < 4; ++i) Dinner *= dims[i];
    int Douter = 1; for (int i = 0; i < k; ++i) Douter *= dims[i];
    int R = ranks[k];
    int total = Douter * R * Dinner;
    mode_dot_kernel<<<(total + 255) / 256, 256, 0, stream>>>(T, Ufac[k], Out,
                                                             Dk, Dinner, R, total);
    dims[k] = R;
  };

  // HOSVD init: per-mode Gram + subspace
  {
    const int dims0[4] = {32, 100, 128, 64};
    for (int m = 0; m < 4; ++m) { gram(Y, dims0, m); eig(m, dims0[m]); }
  }

  // HOOI sweeps
  for (int it = 0; it < 5; ++it) {
    for (int m = 0; m < 4; ++m) {
      int dims[4] = {32, 100, 128, 64};
      const float* cur = Y;
      float* bufs[2] = {BA, BB};
      int pb = 0;
      for (int k = 0; k < 4; ++k) {
        if (k == m) continue;
        modedot(cur, dims, k, bufs[pb]);
        cur = bufs[pb];
        pb ^= 1;
      }
      gram(cur, dims, m);
      eig(m, dims[m]);
    }
  }

  // core = t x0 U0^T x1 U1^T x2 U2^T x3 U3^T  -> d_out (1728 floats)
  {
    int dims[4] = {32, 100, 128, 64};
    const float* cur = Y;
    float* bufs[2] = {BA, BB};
    int pb = 0;
    for (int k = 0; k < 4; ++k) {
      float* dest = (k == 3) ? out : bufs[pb];
      modedot(cur, dims, k, dest);
      cur = dest;
      pb ^= 1;
    }
  }
}